// GlobalAttentionLayer_33011118637646
// MI455X (gfx1250) — compile-verified
//
#include <hip/hip_runtime.h>
#include <hip/hip_bf16.h>

// Problem constants from the reference
#define BB 2
#define TT 512
#define DD 256

// K1 tiling
#define TI 64
#define TJ 64
#define KD 32

#if __has_builtin(__builtin_amdgcn_tanhf)
__device__ __forceinline__ float fast_tanh(float x) { return __builtin_amdgcn_tanhf(x); }
#else
__device__ __forceinline__ float fast_tanh(float x) { return tanhf(x); }
#endif

typedef __attribute__((ext_vector_type(2))) float v2f;
typedef __attribute__((ext_vector_type(8))) float v8f;

// ---------------------------------------------------------------------------
// K1: scores[b,i,j] = sum_d wa[d] * tanh(x[b,i,d]*w1[i,d] + x[b,j,d]*w2[j,d])
// 64x64 output tile per 256-thread block, d staged in 32-wide LDS chunks,
// 4x4 register micro-tile per thread (16 tanh per 8 LDS reads).
// ---------------------------------------------------------------------------
__global__ __launch_bounds__(256) void k_scores(const float* __restrict__ x,
                                                const float* __restrict__ w1,
                                                const float* __restrict__ w2,
                                                const float* __restrict__ wa,
                                                float* __restrict__ scores)
{
    __shared__ float As[TI][KD + 1];   // +1 pad: conflict-free column reads
    __shared__ float Cs[TJ][KD + 1];
    __shared__ float Ws[KD];

    const int blk = blockIdx.x;                 // B * (T/TI) * (T/TJ) = 2*8*8 = 128
    const int b   = blk >> 6;
    const int rem = blk & 63;
    const int i0  = (rem >> 3) * TI;
    const int j0  = (rem & 7) * TJ;

    const int tid = threadIdx.x;
    const int tj  = tid & 15;                   // 16x16 thread grid
    const int ti  = tid >> 4;

    float acc[4][4];
#pragma unroll
    for (int r = 0; r < 4; ++r)
#pragma unroll
        for (int c = 0; c < 4; ++c) acc[r][c] = 0.0f;

    const float* xb = x + (size_t)b * TT * DD;

    for (int d0 = 0; d0 < DD; d0 += KD) {
        // Stage A-tile (i rows) and C-tile (j rows), computing x*w on the fly.
        // Consecutive threads -> consecutive d: coalesced global loads.
#pragma unroll
        for (int k = tid; k < TI * KD; k += 256) {
            const int row = k >> 5;             // KD == 32
            const int dd  = k & (KD - 1);
            const size_t ga = (size_t)(i0 + row) * DD + d0 + dd;
            const size_t gc = (size_t)(j0 + row) * DD + d0 + dd;
            As[row][dd] = xb[ga] * w1[ga];
            Cs[row][dd] = xb[gc] * w2[gc];
        }
        if (tid < KD) Ws[tid] = wa[d0 + tid];
        __syncthreads();

#pragma unroll 4
        for (int dd = 0; dd < KD; ++dd) {
            const float w = Ws[dd];
            float av[4], cv[4];
#pragma unroll
            for (int r = 0; r < 4; ++r) av[r] = As[ti * 4 + r][dd];
#pragma unroll
            for (int c = 0; c < 4; ++c) cv[c] = Cs[tj * 4 + c][dd];
#pragma unroll
            for (int r = 0; r < 4; ++r)
#pragma unroll
                for (int c = 0; c < 4; ++c)
                    acc[r][c] = fmaf(w, fast_tanh(av[r] + cv[c]), acc[r][c]);
        }
        __syncthreads();
    }

    float* sb = scores + (size_t)b * TT * TT;
#pragma unroll
    for (int r = 0; r < 4; ++r)
#pragma unroll
        for (int c = 0; c < 4; ++c)
            sb[(size_t)(i0 + ti * 4 + r) * TT + (j0 + tj * 4 + c)] = acc[r][c];
}

// ---------------------------------------------------------------------------
// K2: in-place row softmax over j (axis 2). One 256-thread block per (b,i) row.
// ---------------------------------------------------------------------------
__global__ __launch_bounds__(256) void k_softmax(float* __restrict__ s)
{
    __shared__ float red[256];
    float* p = s + (size_t)blockIdx.x * TT;
    const int tid = threadIdx.x;

    const float v0 = p[tid];
    const float v1 = p[tid + 256];

    red[tid] = fmaxf(v0, v1);
    __syncthreads();
    for (int off = 128; off > 0; off >>= 1) {
        if (tid < off) red[tid] = fmaxf(red[tid], red[tid + off]);
        __syncthreads();
    }
    const float m = red[0];
    __syncthreads();

    const float e0 = __expf(v0 - m);
    const float e1 = __expf(v1 - m);
    red[tid] = e0 + e1;
    __syncthreads();
    for (int off = 128; off > 0; off >>= 1) {
        if (tid < off) red[tid] += red[tid + off];
        __syncthreads();
    }
    const float inv = 1.0f / red[0];
    p[tid]       = e0 * inv;
    p[tid + 256] = e1 * inv;
}

// ---------------------------------------------------------------------------
// K3: col[b,j] = sum_i normed[b,i,j]  ==  ones(1xT) x normed(TxT).
// One wave (32 lanes) per 16-column strip; V_WMMA_F32_16X16X4_F32 with an
// all-ones A accumulates 4 rows of B per issue into the f32 accumulator.
// Under a ones-A, D[m,n] = sum_k B[k,n] for every m, so any bijection of
// (k,n) onto lanes gives the exact column sum; lanes map n = lane&15.
// ---------------------------------------------------------------------------
__global__ __launch_bounds__(32) void k_colsum(const float* __restrict__ normed,
                                               float* __restrict__ col)
{
    const int strip = blockIdx.x;               // B*(T/16) = 64
    const int b     = strip / (TT / 16);
    const int jbase = (strip % (TT / 16)) * 16;
    const int lane  = threadIdx.x;

    const float* nb = normed + (size_t)b * TT * TT;
    const int j    = jbase + (lane & 15);
    const int rofs = (lane >> 4) << 1;          // lanes 0-15: rows k=0,1; 16-31: k=2,3

    v2f a; a.x = 1.0f; a.y = 1.0f;              // ones A-matrix (16x4)
    v8f c = {};

    for (int i0 = 0; i0 < TT; i0 += 4) {
        v2f bm;
        bm.x = nb[(size_t)(i0 + rofs)     * TT + j];
        bm.y = nb[(size_t)(i0 + rofs + 1) * TT + j];
        // 8 args: (neg_a, A, neg_b, B, c_mod, C, reuse_a, reuse_b)
        c = __builtin_amdgcn_wmma_f32_16x16x4_f32(false, a, false, bm,
                                                  (short)0, c, false, false);
    }
    if (lane < 16) col[(size_t)b * TT + jbase + lane] = c[0];
}

// ---------------------------------------------------------------------------
// K4: out[b,t,0:D] = x[b,t,:]*col[b,t];  out[b,t,D:2D] = x[b,t,:]
// ---------------------------------------------------------------------------
__global__ __launch_bounds__(256) void k_out(const float* __restrict__ x,
                                             const float* __restrict__ col,
                                             float* __restrict__ out)
{
    const int idx = blockIdx.x * 256 + threadIdx.x;
    if (idx >= BB * TT * 2 * DD) return;
    const int d = idx % (2 * DD);
    const int t = (idx / (2 * DD)) % TT;
    const int b = idx / (2 * DD * TT);
    const float xv = x[((size_t)b * TT + t) * DD + (d < DD ? d : d - DD)];
    out[idx] = (d < DD) ? xv * col[(size_t)b * TT + t] : xv;
}

// ---------------------------------------------------------------------------
extern "C" void kernel_launch(void* const* d_in, const int* in_sizes, int n_in,
                              void* d_out, int out_size, void* d_ws, size_t ws_size,
                              hipStream_t stream)
{
    const float* x  = (const float*)d_in[0];   // [B,T,D]
    const float* w1 = (const float*)d_in[1];   // [T,D]
    const float* w2 = (const float*)d_in[2];   // [T,D]
    const float* wa = (const float*)d_in[3];   // [D]
    float* out = (float*)d_out;                // [B,T,2D]

    // Workspace layout: scores/normed [B,T,T] then col [B,T]  (~2 MB + 4 KB)
    float* scores = (float*)d_ws;
    float* col    = scores + (size_t)BB * TT * TT;

    k_scores <<<BB * (TT / TI) * (TT / TJ), 256, 0, stream>>>(x, w1, w2, wa, scores);
    k_softmax<<<BB * TT,                    256, 0, stream>>>(scores);
    k_colsum <<<BB * (TT / 16),              32, 0, stream>>>(scores, col);
    k_out    <<<(BB * TT * 2 * DD + 255) / 256, 256, 0, stream>>>(x, col, out);
}